// edgeConv_26353919328931
// MI455X (gfx1250) — compile-verified
//
#include <hip/hip_runtime.h>

typedef __attribute__((ext_vector_type(2))) float v2f;
typedef __attribute__((ext_vector_type(4))) float v4f;
typedef __attribute__((ext_vector_type(8))) float v8f;

#define BB   8
#define FIN  64
#define NPTS 4096
#define FOUT 64
#define KNN  20

#define MT   32          // query rows per workgroup
#define CT   128         // candidate cols per chunk
#define AST  66          // A_lds row stride: even pad -> aligned b64 frag reads
#define BTS  68          // Bt_lds row stride: aligned + conflict-free b64 frags
#define DSS  132         // dist_lds row stride: 16B-aligned b128 selector reads
#define AGS  68          // aggregation staging stride (aliases dist_lds)

#define INFV 3.0e38f

// ---------------- Kernel 1: projections + squared norms --------------------
// y1[b][n][o] = sum_f x[b][f][n] * (W[o][f] - W[o][64+f])
// y2[b][n][o] = sum_f x[b][f][n] * W[o][64+f]
// sq[b][n]    = sum_f x[b][f][n]^2
__global__ __launch_bounds__(256)
void proj_kernel(const float* __restrict__ x, const float* __restrict__ W,
                 float* __restrict__ y1, float* __restrict__ y2,
                 float* __restrict__ sq) {
    __shared__ float xt[64][FIN + 1];
    __shared__ float wl[FOUT][2 * FIN];
    const int b   = blockIdx.y;
    const int n0  = blockIdx.x * 64;
    const int tid = threadIdx.x;
    const float* xb = x + (size_t)b * FIN * NPTS;

    for (int i = tid; i < 64 * FIN; i += 256) {
        int f = i >> 6, t = i & 63;
        xt[t][f] = xb[(size_t)f * NPTS + n0 + t];
    }
    for (int i = tid; i < FOUT * 2 * FIN; i += 256)
        wl[i >> 7][i & 127] = W[i];
    __syncthreads();

    if (tid < 64) {
        float s = 0.f;
        #pragma unroll 8
        for (int f = 0; f < FIN; ++f) { float v = xt[tid][f]; s += v * v; }
        sq[(size_t)b * NPTS + n0 + tid] = s;
    }
    for (int i = tid; i < 64 * FOUT; i += 256) {
        int t = i >> 6, o = i & 63;
        float a1 = 0.f, a2 = 0.f;
        #pragma unroll 8
        for (int f = 0; f < FIN; ++f) {
            float xv = xt[t][f];
            float w2 = wl[o][FIN + f];
            a1 = fmaf(xv, wl[o][f] - w2, a1);
            a2 = fmaf(xv, w2, a2);
        }
        size_t base = ((size_t)b * NPTS + n0 + t) * FOUT + o;
        y1[base] = a1;
        y2[base] = a2;
    }
}

// ---------------- Kernel 2: WMMA distances + parallel top-k + aggregate ----
__global__ __launch_bounds__(256)
void knn_agg_kernel(const float* __restrict__ x,
                    const float* __restrict__ y1, const float* __restrict__ y2,
                    const float* __restrict__ sq, const float* __restrict__ bias,
                    float* __restrict__ out) {
    __shared__ float A_lds[MT * AST];       // 32 query rows x 64 feats
    __shared__ float Bt_lds[CT * BTS];      // 128 candidates x 64 feats (transposed)
    __shared__ float dist_lds[MT * DSS];    // 32 x 128 distances; reused as staging
    __shared__ float sqA[MT];
    __shared__ float sqB[CT];
    __shared__ int   topi[MT][KNN];

    const int b     = blockIdx.y;
    const int m0    = blockIdx.x * MT;
    const int tid   = threadIdx.x;
    const int wv    = tid >> 5;             // wave -> 16-col sub-tile
    const int lane  = tid & 31;
    const int lhalf = lane >> 4;
    const int lmod  = lane & 15;
    const int row   = tid >> 3;             // selector row (0..31)
    const int sub   = tid & 7;              // selector sub-lane (8 per row)
    const int fb    = tid >> 5;             // chunk-load feature base (0..7)
    const int cq    = (tid & 31) * 4;       // chunk-load col (x4 vectorized)
    const float* xb  = x  + (size_t)b * FIN * NPTS;
    const float* sqb = sq + (size_t)b * NPTS;

    for (int i = tid; i < MT * FIN; i += 256) {
        int f = i >> 5, m = i & (MT - 1);
        A_lds[m * AST + f] = xb[(size_t)f * NPTS + m0 + m];
    }
    if (tid < MT) sqA[tid] = sqb[m0 + tid];

    // software pipeline: preload chunk 0 into registers (b128 loads)
    v4f pre[8];
    #pragma unroll
    for (int q = 0; q < 8; ++q)
        pre[q] = *(const v4f*)&xb[(size_t)(fb + 8 * q) * NPTS + cq];
    float sqpre = (tid < CT) ? sqb[tid] : 0.f;

    __syncthreads();

    // Hoist A fragments (chunk-invariant): 16 per row-tile, 2 row-tiles
    v2f af0[16], af1[16];
    #pragma unroll
    for (int k4 = 0; k4 < 16; ++k4) {
        int kb = k4 * 4 + 2 * lhalf;
        af0[k4] = *(const v2f*)&A_lds[lmod * AST + kb];
        af1[k4] = *(const v2f*)&A_lds[(16 + lmod) * AST + kb];
    }

    // private streaming top-20 per selector thread (register-resident)
    float tv[KNN]; int ti[KNN];
    #pragma unroll
    for (int j = 0; j < KNN; ++j) { tv[j] = INFV; ti[j] = 0; }
    float wmax = INFV; int wpos = 0;

    for (int c0 = 0; c0 < NPTS; c0 += CT) {
        // (a) commit prefetched chunk to LDS, transposed
        #pragma unroll
        for (int q = 0; q < 8; ++q) {
            int f = fb + 8 * q;
            Bt_lds[(cq + 0) * BTS + f] = pre[q][0];
            Bt_lds[(cq + 1) * BTS + f] = pre[q][1];
            Bt_lds[(cq + 2) * BTS + f] = pre[q][2];
            Bt_lds[(cq + 3) * BTS + f] = pre[q][3];
        }
        if (tid < CT) sqB[tid] = sqpre;
        __syncthreads();

        // (b) issue next chunk's global loads (latency hidden under compute)
        if (c0 + CT < NPTS) {
            #pragma unroll
            for (int q = 0; q < 8; ++q)
                pre[q] = *(const v4f*)&xb[(size_t)(fb + 8 * q) * NPTS + c0 + CT + cq];
            if (tid < CT) sqpre = sqb[c0 + CT + tid];
        }

        // (c) two 16x16 Gram tiles per wave, K=64: 32 WMMA, B frag shared
        v8f acc0 = {}, acc1 = {};
        const int col = wv * 16 + lmod;
        #pragma unroll
        for (int k4 = 0; k4 < 16; ++k4) {
            int kb = k4 * 4 + 2 * lhalf;
            v2f bv = *(const v2f*)&Bt_lds[col * BTS + kb];       // ds_load_b64
            acc0 = __builtin_amdgcn_wmma_f32_16x16x4_f32(false, af0[k4], false, bv,
                                                         (short)0, acc0, false, false);
            acc1 = __builtin_amdgcn_wmma_f32_16x16x4_f32(false, af1[k4], false, bv,
                                                         (short)0, acc1, false, false);
        }
        #pragma unroll
        for (int r = 0; r < 8; ++r) {
            int m = r + 8 * lhalf;
            float d0 = fmaf(-2.0f, acc0[r], sqA[m] + sqB[col]);
            float d1 = fmaf(-2.0f, acc1[r], sqA[16 + m] + sqB[col]);
            if (c0 + col == m0 + m)      d0 = INFV;   // mask self
            if (c0 + col == m0 + 16 + m) d1 = INFV;
            dist_lds[m * DSS + col]        = d0;
            dist_lds[(16 + m) * DSS + col] = d1;
        }
        __syncthreads();

        // (d) parallel top-k: 8 lanes/row, contiguous 16-candidate slice each,
        //     fetched with 4x ds_load_b128 then compared from registers
        float dc[16];
        #pragma unroll
        for (int q = 0; q < 4; ++q)
            *(v4f*)&dc[4 * q] = *(const v4f*)&dist_lds[row * DSS + sub * 16 + 4 * q];
        #pragma unroll 4
        for (int i2 = 0; i2 < 16; ++i2) {
            float d = dc[i2];
            if (d < wmax) {
                int cg = c0 + sub * 16 + i2;
                #pragma unroll
                for (int j = 0; j < KNN; ++j) {        // cndmask insert
                    bool s = (j == wpos);
                    tv[j] = s ? d : tv[j];
                    ti[j] = s ? cg : ti[j];
                }
                wmax = tv[0]; wpos = 0;
                #pragma unroll
                for (int j = 1; j < KNN; ++j)
                    if (tv[j] > wmax) { wmax = tv[j]; wpos = j; }
            }
        }
        __syncthreads();
    }

    // Merge 8 partial lists per row -> global top-20 via 8-lane shfl reduce
    for (int t = 0; t < KNN; ++t) {
        float mval = tv[0]; int mpos = 0;
        #pragma unroll
        for (int j = 1; j < KNN; ++j)
            if (tv[j] < mval) { mval = tv[j]; mpos = j; }
        int mgid = 0;
        #pragma unroll
        for (int j = 0; j < KNN; ++j) if (j == mpos) mgid = ti[j];

        float v = mval; int g = mgid;
        #pragma unroll
        for (int msk = 1; msk < 8; msk <<= 1) {
            float ov = __shfl_xor(v, msk, 8);
            int   og = __shfl_xor(g, msk, 8);
            if (ov < v || (ov == v && og < g)) { v = ov; g = og; }
        }
        if (mval == v && mgid == g) {           // unique owner consumes
            #pragma unroll
            for (int j = 0; j < KNN; ++j) if (j == mpos) tv[j] = INFV;
        }
        if (sub == 0) topi[row][t] = g;
    }
    __syncthreads();

    // Phase 1: o-consecutive threads -> coalesced b128 gathers of y2 rows;
    // stage into LDS (reuse dist buffer, stride AGS keeps 16B alignment)
    float* agg = dist_lds;
    const float* y2b = y2 + (size_t)b * NPTS * FOUT;
    for (int i = tid; i < MT * 16; i += 256) {
        int m = i >> 4, oq = (i & 15) * 4;
        v4f mx = {-INFV, -INFV, -INFV, -INFV};
        #pragma unroll
        for (int j = 0; j < KNN; ++j) {
            int g = topi[m][j];
            v4f yv = *(const v4f*)&y2b[(size_t)g * FOUT + oq];
            #pragma unroll
            for (int u = 0; u < 4; ++u) mx[u] = fmaxf(mx[u], yv[u]);
        }
        v4f y1v = *(const v4f*)&y1[((size_t)b * NPTS + m0 + m) * FOUT + oq];
        v4f bv4 = *(const v4f*)&bias[oq];
        *(v4f*)&agg[m * AGS + oq] = y1v + bv4 + mx;
    }
    __syncthreads();

    // Phase 2: m-consecutive threads -> coalesced output stores (B, Fout, N)
    for (int i = tid; i < MT * FOUT; i += 256) {
        int o = i >> 5, m = i & 31;
        out[((size_t)b * FOUT + o) * NPTS + m0 + m] = agg[m * AGS + o];
    }
}

extern "C" void kernel_launch(void* const* d_in, const int* in_sizes, int n_in,
                              void* d_out, int out_size, void* d_ws, size_t ws_size,
                              hipStream_t stream) {
    const float* x    = (const float*)d_in[0];   // (8, 64, 4096) f32
    const float* W    = (const float*)d_in[1];   // (64, 128) f32
    const float* bias = (const float*)d_in[2];   // (64,) f32
    // d_in[3] is k == 20 (hardcoded as KNN)
    float* out = (float*)d_out;                  // (8, 64, 4096) f32
    float* ws  = (float*)d_ws;
    float* y1  = ws;                                        // 8 MB
    float* y2  = ws + (size_t)BB * NPTS * FOUT;             // 8 MB
    float* sq  = y2 + (size_t)BB * NPTS * FOUT;             // 128 KB
    // total workspace needed: (2*B*N*Fout + B*N)*4 ~= 16.1 MB

    proj_kernel<<<dim3(NPTS / 64, BB), 256, 0, stream>>>(x, W, y1, y2, sq);
    knn_agg_kernel<<<dim3(NPTS / MT, BB), 256, 0, stream>>>(x, y1, y2, sq, bias, out);
}